// fusion_layer_45569603010818
// MI455X (gfx1250) — compile-verified
//
#include <hip/hip_runtime.h>

// Rank-6 WMMA-fused dynamic-depthwise-conv + dual pooling for MI455X (gfx1250).
// fused_pre[f,g] = sum_{r<6} P[f,r]*Q[g,r]  computed with two chained
// v_wmma_f32_16x16x4_f32 per 16x16 tile; lrelu + row/col mean fully fused.
// Branch-free inner loop: uniform ds_load_2addr B-fragments, max-based lrelu,
// per-half column accumulator rows (no EXEC churn, no FP atomics).

typedef float v2f __attribute__((ext_vector_type(2)));
typedef float v8f __attribute__((ext_vector_type(8)));

#define BATCH 128
#define FEAT  512
#define INV_F (1.0f / 512.0f)

__global__ __launch_bounds__(256) void fused_dynconv_pool_kernel(
    const float* __restrict__ arm, const float* __restrict__ x86,
    const float* __restrict__ bet, const float* __restrict__ Wde,
    const float* __restrict__ bde, float* __restrict__ out)
{
    __shared__ float apad[516];         // a[f-1], zero padded
    __shared__ float aapad[516];        // |a[f-1]|
    __shared__ float xpad[516];         // x[g-1], zero padded
    __shared__ float Qs[8][FEAT];       // six filtered column vectors + 2 zero rows
    __shared__ float kk[9];             // dynamic 3x3 kernel
    __shared__ float kkpart[8][9];      // per-wave partial dots
    __shared__ float rowsum[FEAT];
    __shared__ float colsumW[16][FEAT]; // per-(wave,half) column sums

    const int b     = blockIdx.x;
    const int t     = threadIdx.x;
    const int wv    = t >> 5;
    const int lane  = t & 31;
    const int laneM = lane & 15;
    const bool lo   = (lane < 16);

    const float* a = arm + b * FEAT;
    const float* x = x86 + b * FEAT;

    // ---- Phase 1: padded loads + zero accumulators ----
    for (int i = t; i < FEAT + 2; i += 256) {
        float av = (i >= 1 && i <= FEAT) ? a[i - 1] : 0.0f;
        float xv = (i >= 1 && i <= FEAT) ? x[i - 1] : 0.0f;
        apad[i]  = av;
        aapad[i] = fabsf(av);
        xpad[i]  = xv;
    }
    for (int i = t; i < 16 * FEAT; i += 256)
        (&colsumW[0][0])[i] = 0.0f;

    // ---- Phase 2: dynamic kernel kk[9] = lrelu(bet . Wde^T + bde) ----
    {
        const float bx = bet[b * FEAT + t];
        const float by = bet[b * FEAT + t + 256];
        for (int j = 0; j < 9; ++j) {
            float p = bx * Wde[j * FEAT + t] + by * Wde[j * FEAT + t + 256];
            p += __shfl_xor(p, 1, 32);
            p += __shfl_xor(p, 2, 32);
            p += __shfl_xor(p, 4, 32);
            p += __shfl_xor(p, 8, 32);
            p += __shfl_xor(p, 16, 32);
            if (lane == 0) kkpart[wv][j] = p;
        }
    }
    __syncthreads();
    if (t < 9) {
        float s = bde[t];
        for (int w = 0; w < 8; ++w) s += kkpart[w][t];   // fixed order
        kk[t] = fmaxf(s, 0.1f * s);                      // lrelu
    }
    __syncthreads();

    // ---- Phase 3: Q vectors (1-D filtered x and |x|, scales folded in) ----
    for (int g = t; g < FEAT; g += 256) {
        const float x0 = xpad[g], x1 = xpad[g + 1], x2 = xpad[g + 2];
        const float u0 = fabsf(x0), u1 = fabsf(x1), u2 = fabsf(x2);
        for (int r = 0; r < 3; ++r) {
            const float k0 = kk[r * 3 + 0], k1 = kk[r * 3 + 1], k2 = kk[r * 3 + 2];
            Qs[r][g]     = 0.55f * (k0 * x0 + k1 * x1 + k2 * x2);
            Qs[r + 3][g] = 0.45f * (k0 * u0 + k1 * u1 + k2 * u2);
        }
        Qs[6][g] = 0.0f;   // zero K-padding rows
        Qs[7][g] = 0.0f;
    }
    __syncthreads();

    // Per-lane K-row bases for the two WMMAs (branch-free B fragments)
    const int r0 = lo ? 0 : 2;   // WMMA1: K0,K1 from rows r0, r0+1
    const int r1 = lo ? 4 : 6;   // WMMA2: K0,K1 from rows r1, r1+1 (6,7 are zero)
    float* colrow = &colsumW[wv * 2 + (lane >> 4)][0];

    // ---- Phase 4: WMMA tile loop (wave owns 4 row-strips of 16 rows) ----
    for (int s = 0; s < 4; ++s) {
        const int f0 = (wv + s * 8) * 16;
        const int f  = f0 + laneM;

        // A fragments (16x4): lanes 0-15 carry K0,K1; lanes 16-31 carry K2,K3
        const float p0 = apad[f],  p1 = apad[f + 1],  p2 = apad[f + 2];
        const float q0 = aapad[f], q1 = aapad[f + 1], q2 = aapad[f + 2];
        v2f afrag1, afrag2;
        afrag1[0] = lo ? p0 : p2;
        afrag1[1] = lo ? p1 : q0;
        afrag2[0] = lo ? q1 : 0.0f;
        afrag2[1] = lo ? q2 : 0.0f;

        v8f rowacc = {};
        for (int gt = 0; gt < 32; ++gt) {
            const int n = gt * 16 + laneM;
            v2f bfrag1, bfrag2;
            bfrag1[0] = Qs[r0][n];     bfrag1[1] = Qs[r0 + 1][n];
            bfrag2[0] = Qs[r1][n];     bfrag2[1] = Qs[r1 + 1][n];

            v8f acc = {};
            acc = __builtin_amdgcn_wmma_f32_16x16x4_f32(
                false, afrag1, false, bfrag1, (short)0, acc, false, false);
            acc = __builtin_amdgcn_wmma_f32_16x16x4_f32(
                false, afrag2, false, bfrag2, (short)0, acc, false, false);

            float colv = 0.0f;
#pragma unroll
            for (int m = 0; m < 8; ++m) {
                const float d = fmaxf(acc[m], 0.1f * acc[m]);  // lrelu
                rowacc[m] += d;
                colv += d;
            }
            colrow[n] += colv;   // unique address per lane, branch-free
        }

        // reduce row accumulators across the 16-lane halves
#pragma unroll
        for (int m = 0; m < 8; ++m) {
            float r = rowacc[m];
            r += __shfl_xor(r, 1, 32);
            r += __shfl_xor(r, 2, 32);
            r += __shfl_xor(r, 4, 32);
            r += __shfl_xor(r, 8, 32);
            if (lane == 0)       rowsum[f0 + m]     = r;
            else if (lane == 16) rowsum[f0 + 8 + m] = r;
        }
    }
    __syncthreads();

    // ---- Phase 5: finalize means and write (B, 2F) output ----
    float* ob = out + b * (2 * FEAT);
    for (int g = t; g < FEAT; g += 256) {
        float cs = 0.0f;
        for (int w = 0; w < 16; ++w) cs += colsumW[w][g];  // fixed order
        ob[g]        = rowsum[g] * INV_F;  // w_pool: mean over columns
        ob[FEAT + g] = cs * INV_F;         // h_pool: mean over rows
    }
}

extern "C" void kernel_launch(void* const* d_in, const int* in_sizes, int n_in,
                              void* d_out, int out_size, void* d_ws, size_t ws_size,
                              hipStream_t stream) {
    (void)in_sizes; (void)n_in; (void)d_ws; (void)ws_size; (void)out_size;
    const float* arm = (const float*)d_in[0];
    const float* x86 = (const float*)d_in[1];
    const float* bet = (const float*)d_in[2];
    const float* Wde = (const float*)d_in[3];
    const float* bde = (const float*)d_in[4];
    float* out = (float*)d_out;

    fused_dynconv_pool_kernel<<<dim3(BATCH), dim3(256), 0, stream>>>(
        arm, x86, bet, Wde, bde, out);
}